// GAT_GCN_Model_82557861363802
// MI455X (gfx1250) — compile-verified
//
#include <hip/hip_runtime.h>
#include <math.h>

#define N_NODES   10000
#define E_EDGES   320000
#define IN_CH     128
#define HID       64
#define HEADS     8
#define HC        512      /* HEADS*HID */
#define OUT_CH    64
#define NEG_SLOPE 0.2f

typedef __attribute__((ext_vector_type(2))) float v2f;
typedef __attribute__((ext_vector_type(8))) float v8f;

// ---------------------------------------------------------------------------
// Monotone float<->uint mapping so atomicMax on u32 implements float max.
// ---------------------------------------------------------------------------
__device__ __forceinline__ unsigned fflip(float f) {
    unsigned u = __float_as_uint(f);
    return (u & 0x80000000u) ? ~u : (u | 0x80000000u);
}
__device__ __forceinline__ float funflip(unsigned u) {
    return __uint_as_float((u & 0x80000000u) ? (u & 0x7FFFFFFFu) : ~u);
}
__device__ __forceinline__ float elu1(float v) {
    return v > 0.f ? v : (__expf(v) - 1.f);
}

// ---------------------------------------------------------------------------
// WMMA fp32 GEMM: C[M,Nc] = A[M,K] @ B[K,Nc] (+bias, +ELU optional).
// One wave32 computes one 16x16 tile of C via K/4 chained v_wmma_f32_16x16x4_f32.
// A 16x4 tile layout: lanes 0-15 hold K={k,k+1}, lanes 16-31 hold K={k+2,k+3}.
// C layout: VGPR r -> row (r + 8*(lane>=16)), col = lane&15.
// Requires M%16==0, Nc%16==0, K%4==0 (all true for this model).
// ---------------------------------------------------------------------------
__global__ __launch_bounds__(128)
void gemm_wmma_f32(const float* __restrict__ A, const float* __restrict__ B,
                   const float* __restrict__ bias, float* __restrict__ C,
                   int M, int K, int Nc, int addBias, int applyElu)
{
    const int lane  = threadIdx.x & 31;
    const int wIdx  = blockIdx.x * (blockDim.x >> 5) + (threadIdx.x >> 5);
    const int tilesN = Nc >> 4;
    const int tileM  = wIdx / tilesN;
    const int tileN  = wIdx % tilesN;
    if (tileM * 16 >= M) return;   // uniform per wave -> EXEC all-ones inside

    const int row0 = tileM * 16, col0 = tileN * 16;
    const int half = lane >> 4;      // 0: K pair {0,1}; 1: K pair {2,3}
    const int mr   = lane & 15;

    v8f c = {};
    const float* __restrict__ Arow = A + (size_t)(row0 + mr) * K;
    for (int k0 = 0; k0 < K; k0 += 4) {
        const int ka = k0 + half * 2;
        v2f a, b;
        a.x = Arow[ka];
        a.y = Arow[ka + 1];
        b.x = B[(size_t)ka       * Nc + col0 + mr];
        b.y = B[(size_t)(ka + 1) * Nc + col0 + mr];
        c = __builtin_amdgcn_wmma_f32_16x16x4_f32(
                /*neg_a=*/false, a, /*neg_b=*/false, b,
                /*c_mod=*/(short)0, c, /*reuse_a=*/false, /*reuse_b=*/false);
    }

    const int   n  = col0 + mr;
    const float bv = addBias ? bias[n] : 0.f;
    const int   m0 = row0 + half * 8;
#pragma unroll
    for (int r = 0; r < 8; ++r) {
        float v = c[r] + bv;
        if (applyElu) v = elu1(v);
        C[(size_t)(m0 + r) * Nc + n] = v;
    }
}

// ---------------------------------------------------------------------------
// Per-(node,head) attention logits: a = <xl[n,h,:], att[h,:]>
// ---------------------------------------------------------------------------
__global__ void gat_attn(const float* __restrict__ xl,
                         const float* __restrict__ att_src,
                         const float* __restrict__ att_dst,
                         float* __restrict__ a_src, float* __restrict__ a_dst)
{
    int idx = blockIdx.x * blockDim.x + threadIdx.x;
    if (idx >= N_NODES * HEADS) return;
    const int n = idx >> 3, h = idx & 7;
    const float* __restrict__ xr = xl + (size_t)n * HC + h * HID;
    const float* __restrict__ as = att_src + h * HID;
    const float* __restrict__ ad = att_dst + h * HID;
    float s0 = 0.f, s1 = 0.f;
#pragma unroll 8
    for (int c = 0; c < HID; ++c) { float v = xr[c]; s0 += v * as[c]; s1 += v * ad[c]; }
    a_src[idx] = s0;
    a_dst[idx] = s1;
}

__global__ void init_mflip(unsigned* __restrict__ m)
{
    int idx = blockIdx.x * blockDim.x + threadIdx.x;
    if (idx < N_NODES * HEADS) m[idx] = fflip(-INFINITY);
}

// ---------------------------------------------------------------------------
// Segment-softmax edge passes. phase 0: atomic max of leaky-relu logit.
// phase 1: atomic add of exp(e - max). One thread per (edge, head).
// Edges e >= E are the implicit self loops (src = dst = e - E).
// ---------------------------------------------------------------------------
__global__ void edge_softmax_pass(const int* __restrict__ srcI, const int* __restrict__ dstI,
                                  const float* __restrict__ a_src, const float* __restrict__ a_dst,
                                  unsigned* __restrict__ mflip, float* __restrict__ ssum,
                                  int phase)
{
    const int total = (E_EDGES + N_NODES) * HEADS;
    for (int idx = blockIdx.x * blockDim.x + threadIdx.x; idx < total;
         idx += gridDim.x * blockDim.x) {
        const int e = idx >> 3, h = idx & 7;
        const int s = (e < E_EDGES) ? srcI[e] : (e - E_EDGES);
        const int d = (e < E_EDGES) ? dstI[e] : (e - E_EDGES);
        float ec = a_src[s * HEADS + h] + a_dst[d * HEADS + h];
        ec = ec > 0.f ? ec : ec * NEG_SLOPE;
        if (phase == 0) {
            atomicMax(&mflip[d * HEADS + h], fflip(ec));
        } else {
            atomicAdd(&ssum[d * HEADS + h], __expf(ec - funflip(mflip[d * HEADS + h])));
        }
    }
}

// ---------------------------------------------------------------------------
// GAT aggregation: one wave32 per edge; lanes stream the 512 contiguous
// channels of xl[src] (fully coalesced) and atomic-add into agg[dst].
// ---------------------------------------------------------------------------
__global__ __launch_bounds__(256)
void gat_aggregate(const int* __restrict__ srcI, const int* __restrict__ dstI,
                   const float* __restrict__ a_src, const float* __restrict__ a_dst,
                   const unsigned* __restrict__ mflip, const float* __restrict__ ssum,
                   const float* __restrict__ xl, float* __restrict__ agg)
{
    const int lane   = threadIdx.x & 31;
    const int wid    = (blockIdx.x * blockDim.x + threadIdx.x) >> 5;
    const int nwaves = (gridDim.x * blockDim.x) >> 5;
    const int total  = E_EDGES + N_NODES;
    for (int e = wid; e < total; e += nwaves) {
        const int s = (e < E_EDGES) ? srcI[e] : (e - E_EDGES);
        const int d = (e < E_EDGES) ? dstI[e] : (e - E_EDGES);
        float alpha = 0.f;
        if (lane < HEADS) {
            float ec = a_src[s * HEADS + lane] + a_dst[d * HEADS + lane];
            ec = ec > 0.f ? ec : ec * NEG_SLOPE;
            alpha = __expf(ec - funflip(mflip[d * HEADS + lane])) / ssum[d * HEADS + lane];
        }
        const float* __restrict__ xs = xl + (size_t)s * HC;
        float* __restrict__ od = agg + (size_t)d * HC;
#pragma unroll
        for (int it = 0; it < HC / 32; ++it) {
            const int i = lane + 32 * it;              // channel, head = i>>6
            const float a = __shfl(alpha, i >> 6);
            atomicAdd(&od[i], a * xs[i]);
        }
    }
}

// ---------------------------------------------------------------------------
// GCN helpers
// ---------------------------------------------------------------------------
__global__ void gcn_degree(const int* __restrict__ dstI, float* __restrict__ deg)
{
    const int total = E_EDGES + N_NODES;
    for (int idx = blockIdx.x * blockDim.x + threadIdx.x; idx < total;
         idx += gridDim.x * blockDim.x) {
        const int d = (idx < E_EDGES) ? dstI[idx] : (idx - E_EDGES);
        atomicAdd(&deg[d], 1.f);
    }
}

__global__ void gcn_dinv(float* __restrict__ deg)
{
    int idx = blockIdx.x * blockDim.x + threadIdx.x;
    if (idx < N_NODES) deg[idx] = rsqrtf(fmaxf(deg[idx], 1e-12f));
}

// One wave32 per edge; 64 contiguous channels (2 iterations).
__global__ __launch_bounds__(256)
void gcn_aggregate(const int* __restrict__ srcI, const int* __restrict__ dstI,
                   const float* __restrict__ dinv,
                   const float* __restrict__ xin, float* __restrict__ out)
{
    const int lane   = threadIdx.x & 31;
    const int wid    = (blockIdx.x * blockDim.x + threadIdx.x) >> 5;
    const int nwaves = (gridDim.x * blockDim.x) >> 5;
    const int total  = E_EDGES + N_NODES;
    for (int e = wid; e < total; e += nwaves) {
        const int s = (e < E_EDGES) ? srcI[e] : (e - E_EDGES);
        const int d = (e < E_EDGES) ? dstI[e] : (e - E_EDGES);
        const float norm = dinv[s] * dinv[d];
        const float* __restrict__ xs = xin + (size_t)s * HID;
        float* __restrict__ od = out + (size_t)d * HID;
#pragma unroll
        for (int it = 0; it < HID / 32; ++it) {
            const int i = lane + 32 * it;
            atomicAdd(&od[i], norm * xs[i]);
        }
    }
}

// In-place bias (+optional ELU) over [total] with channel count C (power of 2).
__global__ void bias_act(float* __restrict__ x, const float* __restrict__ bias,
                         int total, int C, int applyElu)
{
    for (int idx = blockIdx.x * blockDim.x + threadIdx.x; idx < total;
         idx += gridDim.x * blockDim.x) {
        float v = x[idx] + bias[idx & (C - 1)];
        if (applyElu) v = elu1(v);
        x[idx] = v;
    }
}

// ---------------------------------------------------------------------------
// Launch
// ---------------------------------------------------------------------------
extern "C" void kernel_launch(void* const* d_in, const int* in_sizes, int n_in,
                              void* d_out, int out_size, void* d_ws, size_t ws_size,
                              hipStream_t stream)
{
    const float* x       = (const float*)d_in[0];
    const int*   ei      = (const int*)  d_in[1];
    const int*   srcI    = ei;             // edge_index[0]
    const int*   dstI    = ei + E_EDGES;   // edge_index[1]
    const float* W_gat   = (const float*)d_in[2];
    const float* att_src = (const float*)d_in[3];
    const float* att_dst = (const float*)d_in[4];
    const float* b_gat   = (const float*)d_in[5];
    const float* W_emb   = (const float*)d_in[6];
    const float* b_emb   = (const float*)d_in[7];
    const float* W_g1    = (const float*)d_in[8];
    const float* b_g1    = (const float*)d_in[9];
    const float* W_g2    = (const float*)d_in[10];
    const float* b_g2    = (const float*)d_in[11];
    float*       out     = (float*)d_out;

    // Workspace layout (floats)
    float*    ws     = (float*)d_ws;
    float*    xl     = ws;                                // N*512 (reused below)
    float*    agg    = ws + (size_t)N_NODES * HC;         // N*512
    float*    a_src  = agg + (size_t)N_NODES * HC;        // N*8
    float*    a_dst  = a_src + N_NODES * HEADS;           // N*8
    unsigned* mflip  = (unsigned*)(a_dst + N_NODES * HEADS); // N*8
    float*    ssum   = (float*)(mflip + N_NODES * HEADS); // N*8
    // xl region reuse after the GAT scatter (8 slots of N*64 available):
    float* h2   = xl;                                     // N*64
    float* t1   = xl + (size_t)1 * N_NODES * HID;         // N*64
    float* deg  = xl + (size_t)2 * N_NODES * HID;         // N
    float* agg2 = xl + (size_t)3 * N_NODES * HID;         // N*64
    float* t2   = xl + (size_t)4 * N_NODES * HID;         // N*64

    const int edgeThreads  = (E_EDGES + N_NODES) * HEADS;       // 2.64M
    const int edgeBlocks   = (edgeThreads + 255) / 256;
    const int waveBlocks   = ((E_EDGES + N_NODES) * 32 + 255) / 256; // wave per edge
    const int nhBlocks     = (N_NODES * HEADS + 255) / 256;

    // ---- GAT: xl = x @ W_gat  (WMMA f32, 625x32 tiles, 4 waves/block) ----
    gemm_wmma_f32<<<(625 * 32) / 4, 128, 0, stream>>>(
        x, W_gat, nullptr, xl, N_NODES, IN_CH, HC, 0, 0);

    // attention logits + softmax state init
    gat_attn<<<nhBlocks, 256, 0, stream>>>(xl, att_src, att_dst, a_src, a_dst);
    init_mflip<<<nhBlocks, 256, 0, stream>>>(mflip);
    hipMemsetAsync(ssum, 0, (size_t)N_NODES * HEADS * sizeof(float), stream);
    hipMemsetAsync(agg,  0, (size_t)N_NODES * HC    * sizeof(float), stream);

    // segment softmax (max, then sum of exp)
    edge_softmax_pass<<<edgeBlocks, 256, 0, stream>>>(srcI, dstI, a_src, a_dst, mflip, ssum, 0);
    edge_softmax_pass<<<edgeBlocks, 256, 0, stream>>>(srcI, dstI, a_src, a_dst, mflip, ssum, 1);

    // weighted scatter of xl[src] into agg[dst]
    gat_aggregate<<<waveBlocks, 256, 0, stream>>>(srcI, dstI, a_src, a_dst, mflip, ssum, xl, agg);

    // + b_gat, ELU  -> h (in place in agg)
    bias_act<<<(N_NODES * HC + 255) / 256, 256, 0, stream>>>(agg, b_gat, N_NODES * HC, HC, 1);

    // ---- node embedding: h2 = ELU(h @ W_emb + b_emb) ----
    gemm_wmma_f32<<<(625 * 4) / 4, 128, 0, stream>>>(
        agg, W_emb, b_emb, h2, N_NODES, HC, HID, 1, 1);

    // ---- GCN degree normalization (shared by both GCN layers) ----
    hipMemsetAsync(deg, 0, (size_t)N_NODES * sizeof(float), stream);
    gcn_degree<<<(E_EDGES + N_NODES + 255) / 256, 256, 0, stream>>>(dstI, deg);
    gcn_dinv<<<(N_NODES + 255) / 256, 256, 0, stream>>>(deg);

    // ---- GCN layer 1: t1 = h2 @ W_g1 ; agg2 = scatter(norm * t1) ; +b, ELU ----
    gemm_wmma_f32<<<(625 * 4) / 4, 128, 0, stream>>>(
        h2, W_g1, nullptr, t1, N_NODES, HID, HID, 0, 0);
    hipMemsetAsync(agg2, 0, (size_t)N_NODES * HID * sizeof(float), stream);
    gcn_aggregate<<<waveBlocks, 256, 0, stream>>>(srcI, dstI, deg, t1, agg2);
    bias_act<<<(N_NODES * HID + 255) / 256, 256, 0, stream>>>(agg2, b_g1, N_NODES * HID, HID, 1);

    // ---- GCN layer 2: t2 = h3 @ W_g2 ; out = scatter(norm * t2) + b_g2 ----
    gemm_wmma_f32<<<(625 * 4) / 4, 128, 0, stream>>>(
        agg2, W_g2, nullptr, t2, N_NODES, HID, OUT_CH, 0, 0);
    hipMemsetAsync(out, 0, (size_t)N_NODES * OUT_CH * sizeof(float), stream);
    gcn_aggregate<<<waveBlocks, 256, 0, stream>>>(srcI, dstI, deg, t2, out);
    bias_act<<<(N_NODES * OUT_CH + 255) / 256, 256, 0, stream>>>(out, b_g2, N_NODES * OUT_CH, OUT_CH, 0);
}